// CFAdapter_1606317768726
// MI455X (gfx1250) — compile-verified
//
#include <hip/hip_runtime.h>
#include <hip/hip_bf16.h>
#include <math.h>

typedef __attribute__((ext_vector_type(2))) float v2f;
typedef __attribute__((ext_vector_type(8))) float v8f;
typedef __attribute__((ext_vector_type(4))) unsigned int u32x4;
typedef __attribute__((ext_vector_type(4))) int i32x4;
typedef __attribute__((ext_vector_type(8))) int i32x8;

#define D_MODEL 4096
#define FAN_IN1 4112           // D_MODEL + 16
#define N_TOT   80             // 16 fiber cols + 64 control cols
#define MT      64             // rows per block
#define KC      64             // K chunk
#define LDA     68             // LDS stride (pad 4 floats -> bank-conflict free)
#define LDB     68
#define LDX     84             // 80 + 4 pad
#define BUF_F   (MT * LDA + N_TOT * LDB)   // floats per double-buffer half

#if __has_builtin(__builtin_amdgcn_tensor_load_to_lds)
#define USE_TDM 1
#endif

#if defined(USE_TDM)
// ---------------------------------------------------------------------------
// Tensor Data Mover: DMA a rows x 64 fp32 tile (row stride = stride_elems)
// from global into LDS, inserting 4 DWORDs of padding after every 64 DWORDs
// so the LDS image lands with stride LDA/LDB = 68 floats per row.
// D# layout per CDNA5 ISA ch.8 (group0: count/lds/global/type, group1: dims).
// ---------------------------------------------------------------------------
static __device__ __forceinline__
void tdm_load_tile(const float* gsrc, float* ldst, int rows, int stride_elems)
{
    unsigned long long ga = (unsigned long long)(__UINTPTR_TYPE__)gsrc;
    unsigned int lds = (unsigned int)(__UINTPTR_TYPE__)ldst; // low 32b = LDS offset

    u32x4 g0;
    g0.x = 1u;                                   // count=1, is_restore=0
    g0.y = lds;                                  // lds_addr (bytes)
    g0.z = (unsigned int)ga;                     // global_addr[31:0]
    g0.w = (unsigned int)(ga >> 32) | (2u << 30);// global_addr[56:32] | type=2

    i32x8 g1;
    // wg_mask=0 | data_size=4B | pad_enable | pad_interval=64dw | pad_amount=4dw
    g1[0] = (int)((2u << 16) | (1u << 20) | (5u << 22) | (3u << 25));
    g1[1] = (int)(64u << 16);                    // tensor_dim0.lo16 = 64
    g1[2] = (int)((unsigned)rows << 16);         // tensor_dim0.hi=0 | tensor_dim1.lo16
    g1[3] = (int)(64u << 16);                    // tensor_dim1.hi=0 | tile_dim0 = 64
    g1[4] = rows;                                // tile_dim1 = rows | tile_dim2 = 0
    g1[5] = stride_elems;                        // tensor_dim0_stride[31:0]
    g1[6] = 0;                                   // stride hi16 | dim1_stride lo16
    g1[7] = 0;

    i32x4 gz = {0, 0, 0, 0};
#if __clang_major__ >= 23
    i32x8 gz8 = {0, 0, 0, 0, 0, 0, 0, 0};
    __builtin_amdgcn_tensor_load_to_lds(g0, g1, gz, gz, gz8, 0);
#else
    __builtin_amdgcn_tensor_load_to_lds(g0, g1, gz, gz, 0);
#endif
}
#endif

// ---------------------------------------------------------------------------
// Kernel A: fused GEMM  X = h @ [W_fiber; W1[:, :4096]]^T   (8192 x 80)
// TDM double-buffered staging -> f32 WMMA -> per-row epilogue:
// Y = X[:,16:] + X[:,:16] @ W1f^T + b1, z = gelu(Y), delta = softplus(z@W2+b2)
// ---------------------------------------------------------------------------
__global__ __launch_bounds__(128)
void cfadapter_gemm_kernel(const float* __restrict__ hidden,
                           const float* __restrict__ Wf,    // [16,4096]
                           const float* __restrict__ W1,    // [64,4112]
                           const float* __restrict__ b1,    // [64]
                           const float* __restrict__ W2,    // [64]
                           const float* __restrict__ b2,    // [1]
                           float* __restrict__ delta_out,   // [n_rows]
                           int n_rows)
{
    __shared__ float smem[2 * BUF_F];  // double-buffered A|B tiles (reused as X)
    __shared__ float sW1f[64 * 16];
    __shared__ float sB1[64];
    __shared__ float sW2[64];

    const int tid  = threadIdx.x;
    const int lane = tid & 31;
    const int wave = tid >> 5;
    const int row0 = blockIdx.x * MT;

    // preload small weights: W1 fiber columns, b1, W2
    for (int i = tid; i < 64 * 16; i += 128)
        sW1f[i] = W1[(size_t)(i >> 4) * FAN_IN1 + D_MODEL + (i & 15)];
    for (int i = tid; i < 64; i += 128) { sB1[i] = b1[i]; sW2[i] = W2[i]; }

#if defined(USE_TDM)
#define STAGE_CHUNK(buf, k0)                                                   \
    do { if (wave == 0) {                                                      \
        float* dA = &smem[(buf) * BUF_F];                                      \
        float* dB = dA + MT * LDA;                                             \
        tdm_load_tile(&hidden[(size_t)row0 * D_MODEL + (k0)], dA, MT, D_MODEL);\
        tdm_load_tile(&Wf[(k0)], dB, 16, D_MODEL);                             \
        tdm_load_tile(&W1[(k0)], dB + 16 * LDB, 64, FAN_IN1);                  \
    } } while (0)
#else
#define STAGE_CHUNK(buf, k0)                                                   \
    do {                                                                       \
        float* dA = &smem[(buf) * BUF_F];                                      \
        float* dB = dA + MT * LDA;                                             \
        for (int q = tid; q < MT * KC / 4; q += 128) {                         \
            int r = q >> 4, c4 = q & 15;                                       \
            float4 v = *(const float4*)&hidden[(size_t)(row0 + r) * D_MODEL    \
                                               + (k0) + c4 * 4];               \
            float* dst = &dA[r * LDA + c4 * 4];                                \
            dst[0] = v.x; dst[1] = v.y; dst[2] = v.z; dst[3] = v.w;            \
        }                                                                      \
        for (int q = tid; q < N_TOT * KC / 4; q += 128) {                      \
            int n = q >> 4, c4 = q & 15;                                       \
            const float* src = (n < 16)                                        \
                ? &Wf[(size_t)n * D_MODEL + (k0) + c4 * 4]                     \
                : &W1[(size_t)(n - 16) * FAN_IN1 + (k0) + c4 * 4];             \
            float4 v = *(const float4*)src;                                    \
            float* dst = &dB[n * LDB + c4 * 4];                                \
            dst[0] = v.x; dst[1] = v.y; dst[2] = v.z; dst[3] = v.w;            \
        }                                                                      \
    } while (0)
#endif

    v8f acc[5];
#pragma unroll
    for (int j = 0; j < 5; ++j) acc[j] = (v8f)0.0f;

    const int mrow = wave * 16 + (lane & 15);   // LDS row for A fragment
    const int koff = (lane >> 4) * 2;           // lanes 16-31 take K+2,K+3
    const int NCHUNK = D_MODEL / KC;            // 64

    STAGE_CHUNK(0, 0);                          // prologue: chunk 0 in flight

    for (int c = 0; c < NCHUNK; ++c) {
        const int cur = c & 1;
        if (c + 1 < NCHUNK) {
            STAGE_CHUNK(c + 1 < NCHUNK ? ((c + 1) & 1) : 0, (c + 1) * KC);
#if defined(USE_TDM)
            if (wave == 0) __builtin_amdgcn_s_wait_tensorcnt(3); // cur arrived
#endif
        } else {
#if defined(USE_TDM)
            if (wave == 0) __builtin_amdgcn_s_wait_tensorcnt(0);
#endif
        }
        __syncthreads();                        // publish cur chunk to all waves

        const float* sA = &smem[cur * BUF_F];
        const float* sB = sA + MT * LDA;
#pragma unroll 4
        for (int kk = 0; kk < KC; kk += 4) {
            v2f a = *(const v2f*)&sA[mrow * LDA + kk + koff];
#pragma unroll
            for (int j = 0; j < 5; ++j) {
                v2f b = *(const v2f*)&sB[(j * 16 + (lane & 15)) * LDB + kk + koff];
                acc[j] = __builtin_amdgcn_wmma_f32_16x16x4_f32(
                    false, a, false, b, (short)0, acc[j], false, false);
            }
        }
        __syncthreads();                        // all done reading before rewrite
    }

    // spill X tile (64 x 80) to LDS for row-major epilogue
    float* sX = smem;
#pragma unroll
    for (int j = 0; j < 5; ++j)
#pragma unroll
        for (int i = 0; i < 8; ++i)
            sX[(wave * 16 + i + 8 * (lane >> 4)) * LDX + j * 16 + (lane & 15)] = acc[j][i];
    __syncthreads();

    // epilogue: one thread per row
    if (tid < MT) {
        const float* xr = &sX[tid * LDX];
        float xf[16];
#pragma unroll
        for (int f = 0; f < 16; ++f) xf[f] = xr[f];
        float accd = 0.0f;
#pragma unroll 4
        for (int k = 0; k < 64; ++k) {
            float y = xr[16 + k] + sB1[k];
#pragma unroll
            for (int f = 0; f < 16; ++f) y = fmaf(xf[f], sW1f[k * 16 + f], y);
            float g = 0.5f * y * (1.0f + erff(y * 0.70710678118654752f)); // exact gelu
            accd = fmaf(g, sW2[k], accd);
        }
        float t = accd + b2[0];
        float sp = (t > 0.0f) ? t + log1pf(expf(-t)) : log1pf(expf(t)); // softplus
        delta_out[row0 + tid] = sp;
    }
}

// ---------------------------------------------------------------------------
// Kernel B: causal EMA scan (one wave per batch, affine-recurrence wave scan),
// gate = sigmoid(-lam*field), plus mean(delta). out = [gate | field | mean]
// ---------------------------------------------------------------------------
__global__ __launch_bounds__(128)
void cfadapter_ema_kernel(const float* __restrict__ delta,
                          const float* __restrict__ lam,
                          float* __restrict__ out,
                          int Slen, int nBatch, int total)
{
    const int lane = threadIdx.x & 31;
    const int wave = threadIdx.x >> 5;
    const float alpha = 0.9f;
    const float lamv = lam[0];
    float lsum = 0.0f;

    if (wave < nBatch) {
        float carry = 0.0f;
        for (int c = 0; c < Slen; c += 32) {
            int idx = wave * Slen + c + lane;
            float d = delta[idx];
            lsum += d;
            // affine transform h -> A*h + B, inclusive scan over lanes
            float A = alpha;
            float Bv = (1.0f - alpha) * d;
#pragma unroll
            for (int off = 1; off < 32; off <<= 1) {
                float Ap = __shfl_up(A, off, 32);
                float Bp = __shfl_up(Bv, off, 32);
                if (lane >= off) { Bv = fmaf(A, Bp, Bv); A = A * Ap; }
            }
            float field = fmaf(A, carry, Bv);
            carry = __shfl(field, 31, 32);
            float gate = 1.0f / (1.0f + expf(lamv * field)); // sigmoid(-lam*field)
            out[idx] = gate;
            out[total + idx] = field;
        }
    }

    // block reduction for mean(delta)
    __shared__ float red[4];
#pragma unroll
    for (int off = 16; off > 0; off >>= 1) lsum += __shfl_down(lsum, off, 32);
    if (lane == 0) red[wave] = lsum;
    __syncthreads();
    if (threadIdx.x == 0) {
        float s = 0.0f;
        for (int w = 0; w < 4; ++w) s += red[w];
        out[2 * total] = s / (float)total;
    }
}

// ---------------------------------------------------------------------------
extern "C" void kernel_launch(void* const* d_in, const int* in_sizes, int n_in,
                              void* d_out, int out_size, void* d_ws, size_t ws_size,
                              hipStream_t stream) {
    const float* hidden = (const float*)d_in[0];
    const float* Wf     = (const float*)d_in[1];
    const float* W1     = (const float*)d_in[2];
    const float* b1     = (const float*)d_in[3];
    const float* W2     = (const float*)d_in[4];
    const float* b2     = (const float*)d_in[5];
    const float* lam    = (const float*)d_in[6];
    float* out   = (float*)d_out;
    float* delta = (float*)d_ws;    // [n_rows] scratch

    const int n_rows = in_sizes[0] / D_MODEL;   // B*S = 8192
    const int nBatch = 4;                       // per reference setup
    const int S      = n_rows / nBatch;         // 2048

    cfadapter_gemm_kernel<<<dim3(n_rows / MT), 128, 0, stream>>>(
        hidden, Wf, W1, b1, W2, b2, delta, n_rows);
    cfadapter_ema_kernel<<<dim3(1), 128, 0, stream>>>(
        delta, lam, out, S, nBatch, n_rows);
}